// FiberGNN_82970178224656
// MI455X (gfx1250) — compile-verified
//
#include <hip/hip_runtime.h>

#define HID 64
#define INC 32
#define OUTC 16
#define NG 64
#define EPS_LN 1e-5f

typedef __attribute__((ext_vector_type(2))) float v2f;
typedef __attribute__((ext_vector_type(8))) float v8f;

__device__ __forceinline__ float gelu_exact(float x) {
  return 0.5f * x * (1.0f + erff(x * 0.7071067811865475f));
}

__device__ __forceinline__ void atomic_add_f32(float* p, float v) {
  unsafeAtomicAdd(p, v);  // hardware global_atomic_add_f32 on CDNA
}

extern "C" __global__ void k_zero(float* __restrict__ p, long n) {
  long i = (long)blockIdx.x * blockDim.x + threadIdx.x;
  if (i < n) p[i] = 0.0f;
}

extern "C" __global__ void k_deg(const int* __restrict__ dst, const float* __restrict__ ew,
                                 float* __restrict__ deg, int e) {
  int i = blockIdx.x * blockDim.x + threadIdx.x;
  if (i < e) atomic_add_f32(&deg[dst[i]], ew[i]);
}

extern "C" __global__ void k_dis(const float* __restrict__ deg, float* __restrict__ dis, int n) {
  int i = blockIdx.x * blockDim.x + threadIdx.x;
  if (i < n) dis[i] = rsqrtf(deg[i] + 1.0f);  // +1 = self-loop weight
}

// Y[n x 64] = act(X[n x CI] @ W[64 x CI]^T + bias), full f32 via V_WMMA_F32_16X16X4_F32.
// 128 threads = 4 waves; wave w computes rows [blk*64 + w*16, +16) x all 64 cols.
extern "C" __global__ void __launch_bounds__(128)
k_gemm_wmma(const float* __restrict__ X, const float* __restrict__ W,
            const float* __restrict__ bias, float* __restrict__ Y,
            int n, int CI, int do_gelu) {
  __shared__ float ldsW[HID * (HID + 1)];  // padded stride (CI+1) -> conflict-free
  const int tid = threadIdx.x;
  const int stride = CI + 1;
  for (int t = tid; t < HID * CI; t += 128) {
    int j = t / CI, k = t - j * CI;
    ldsW[j * stride + k] = W[t];
  }
  __syncthreads();

  const int wave = tid >> 5;
  const int lane = tid & 31;
  const int half = lane >> 4;  // 0: K=k0,k0+1 ; 1: K=k0+2,k0+3
  const int l    = lane & 15;
  const int row0 = blockIdx.x * 64 + wave * 16;

  v8f acc[4] = {};
  const int arow = row0 + l;
  const float* xrow = X + (size_t)arow * CI;

  for (int k0 = 0; k0 < CI; k0 += 4) {
    const int kk = k0 + half * 2;
    v2f a = {};
    if (arow < n) a = *(const v2f*)(xrow + kk);  // A 16x4 frag: contiguous K pair
#pragma unroll
    for (int jt = 0; jt < 4; ++jt) {
      const float* wp = &ldsW[(jt * 16 + l) * stride + kk];  // B[k][j] = W[j][k]
      v2f b;
      b.x = wp[0];
      b.y = wp[1];
      acc[jt] = __builtin_amdgcn_wmma_f32_16x16x4_f32(
          false, a, false, b, (short)0, acc[jt], false, false);
    }
  }

  const int orow_base = row0 + half * 8;  // C layout: VGPR v -> M = v (+8 for upper lanes)
#pragma unroll
  for (int jt = 0; jt < 4; ++jt) {
    const int col = jt * 16 + l;
    const float bv = bias ? bias[col] : 0.0f;
#pragma unroll
    for (int v = 0; v < 8; ++v) {
      const int row = orow_base + v;
      if (row < n) {
        float t = acc[jt][v] + bv;
        if (do_gelu) t = gelu_exact(t);
        Y[(size_t)row * HID + col] = t;
      }
    }
  }
}

// 16 lanes per edge, float4 per lane: coalesced 256B row gather + HW FP atomics.
extern "C" __global__ void k_scatter(const float* __restrict__ xl, const int* __restrict__ src,
                                     const int* __restrict__ dst, const float* __restrict__ ew,
                                     const float* __restrict__ dis, float* __restrict__ agg,
                                     int e) {
  long idx = (long)blockIdx.x * blockDim.x + threadIdx.x;
  int ei = (int)(idx >> 4);
  if (ei >= e) return;
  int c = ((int)idx & 15) << 2;
  int s = src[ei], d = dst[ei];
  float norm = dis[s] * ew[ei] * dis[d];
  float4 v = *(const float4*)(xl + (size_t)s * HID + c);
  float* o = agg + (size_t)d * HID + c;
  atomic_add_f32(o + 0, v.x * norm);
  atomic_add_f32(o + 1, v.y * norm);
  atomic_add_f32(o + 2, v.z * norm);
  atomic_add_f32(o + 3, v.w * norm);
}

// Fused: self-loop + bias + gelu + layernorm (wave32 shuffle reduce) + residual.
// One wave per node, 2 channels per lane.
extern "C" __global__ void __launch_bounds__(256)
k_post(const float* __restrict__ agg, const float* __restrict__ xl,
       const float* __restrict__ dis, const float* __restrict__ bias,
       const float* __restrict__ gamma, const float* __restrict__ beta,
       float* __restrict__ h, int n) {
  const int wave = threadIdx.x >> 5;
  const int lane = threadIdx.x & 31;
  const int i = blockIdx.x * 8 + wave;
  if (i >= n) return;
  float d2 = dis[i];
  d2 *= d2;  // self-loop norm = dis*1*dis
  const size_t base = (size_t)i * HID;
  const int c0 = lane, c1 = lane + 32;
  float g0 = gelu_exact(agg[base + c0] + xl[base + c0] * d2 + bias[c0]);
  float g1 = gelu_exact(agg[base + c1] + xl[base + c1] * d2 + bias[c1]);
  float s = g0 + g1;
#pragma unroll
  for (int m = 16; m >= 1; m >>= 1) s += __shfl_xor(s, m, 32);
  const float mu = s * (1.0f / 64.0f);
  const float q0 = g0 - mu, q1 = g1 - mu;
  float vs = q0 * q0 + q1 * q1;
#pragma unroll
  for (int m = 16; m >= 1; m >>= 1) vs += __shfl_xor(vs, m, 32);
  const float inv = rsqrtf(vs * (1.0f / 64.0f) + EPS_LN);
  h[base + c0] += q0 * inv * gamma[c0] + beta[c0];
  h[base + c1] += q1 * inv * gamma[c1] + beta[c1];
}

extern "C" __global__ void k_pool(const float* __restrict__ h, const int* __restrict__ batch,
                                  float* __restrict__ psum, float* __restrict__ pcnt, int n) {
  long idx = (long)blockIdx.x * blockDim.x + threadIdx.x;
  int i = (int)(idx >> 6);
  if (i >= n) return;
  int c = (int)idx & 63;
  int b = batch[i];
  atomic_add_f32(&psum[b * HID + c], h[(size_t)i * HID + c]);
  if (c == 0) atomic_add_f32(&pcnt[b], 1.0f);
}

extern "C" __global__ void k_head(const float* __restrict__ psum, const float* __restrict__ pcnt,
                                  const float* __restrict__ pw, const float* __restrict__ pb,
                                  float* __restrict__ out) {
  const int t = threadIdx.x;  // 1024 = 64 graphs x 16 outputs
  const int g = t >> 4, o = t & 15;
  const float inv = 1.0f / fmaxf(pcnt[g], 1.0f);
  float acc = 0.0f;
#pragma unroll 8
  for (int c = 0; c < HID; ++c) acc += psum[g * HID + c] * inv * pw[o * HID + c];
  out[g * OUTC + o] = acc + pb[o];
}

extern "C" void kernel_launch(void* const* d_in, const int* in_sizes, int n_in,
                              void* d_out, int out_size, void* d_ws, size_t ws_size,
                              hipStream_t stream) {
  const float* x      = (const float*)d_in[0];
  const float* ew     = (const float*)d_in[1];
  const float* proj_w = (const float*)d_in[2];
  const float* proj_b = (const float*)d_in[3];
  const float* c1w    = (const float*)d_in[4];
  const float* c1b    = (const float*)d_in[5];
  const float* c2w    = (const float*)d_in[6];
  const float* c2b    = (const float*)d_in[7];
  const float* ln1g   = (const float*)d_in[8];
  const float* ln1b   = (const float*)d_in[9];
  const float* ln2g   = (const float*)d_in[10];
  const float* ln2b   = (const float*)d_in[11];
  const float* pw     = (const float*)d_in[12];
  const float* pb     = (const float*)d_in[13];
  const int*   src    = (const int*)d_in[14];
  const int*   dst    = (const int*)d_in[15];
  const int*   batch  = (const int*)d_in[16];

  const int n = in_sizes[0] / INC;
  const int e = in_sizes[1];

  float* ws   = (float*)d_ws;
  float* deg  = ws;
  float* dis  = deg + n;
  float* h    = dis + n;
  float* xl   = h + (size_t)n * HID;
  float* agg  = xl + (size_t)n * HID;
  float* psum = agg + (size_t)n * HID;
  float* pcnt = psum + NG * HID;

  const long nh = (long)n * HID;
  const int ZB = 256;
  const int gemm_grid = (n + 63) / 64;
  const int post_grid = (n + 7) / 8;
  const int scat_grid = (int)(((long)e * 16 + ZB - 1) / ZB);
  const int pool_grid = (int)((nh + ZB - 1) / ZB);

  // init accumulators (every call — graph replays must be self-contained)
  k_zero<<<(n + ZB - 1) / ZB, ZB, 0, stream>>>(deg, n);
  k_zero<<<(NG * HID + NG + ZB - 1) / ZB, ZB, 0, stream>>>(psum, NG * HID + NG);

  // symmetric-normalization degrees
  k_deg<<<(e + ZB - 1) / ZB, ZB, 0, stream>>>(dst, ew, deg, e);
  k_dis<<<(n + ZB - 1) / ZB, ZB, 0, stream>>>(deg, dis, n);

  // h = gelu(x @ proj_w^T + proj_b)
  k_gemm_wmma<<<gemm_grid, 128, 0, stream>>>(x, proj_w, proj_b, h, n, INC, 1);

  // conv1
  k_gemm_wmma<<<gemm_grid, 128, 0, stream>>>(h, c1w, nullptr, xl, n, HID, 0);
  k_zero<<<(int)((nh + ZB - 1) / ZB), ZB, 0, stream>>>(agg, nh);
  k_scatter<<<scat_grid, ZB, 0, stream>>>(xl, src, dst, ew, dis, agg, e);
  k_post<<<post_grid, 256, 0, stream>>>(agg, xl, dis, c1b, ln1g, ln1b, h, n);

  // conv2
  k_gemm_wmma<<<gemm_grid, 128, 0, stream>>>(h, c2w, nullptr, xl, n, HID, 0);
  k_zero<<<(int)((nh + ZB - 1) / ZB), ZB, 0, stream>>>(agg, nh);
  k_scatter<<<scat_grid, ZB, 0, stream>>>(xl, src, dst, ew, dis, agg, e);
  k_post<<<post_grid, 256, 0, stream>>>(agg, xl, dis, c2b, ln2g, ln2b, h, n);

  // mean pool + head
  k_pool<<<pool_grid, ZB, 0, stream>>>(h, batch, psum, pcnt, n);
  k_head<<<1, 1024, 0, stream>>>(psum, pcnt, pw, pb, (float*)d_out);
}